// SparsePool_59416577573008
// MI455X (gfx1250) — compile-verified
//
#include <hip/hip_runtime.h>

#define D_FEAT 128   // floats per node row; 32 lanes x float4 = one row per wave

// clang native vector type: accepted by __builtin_nontemporal_load/store
typedef float v4f __attribute__((ext_vector_type(4)));

#if defined(__has_builtin)
#if __has_builtin(__builtin_amdgcn_global_load_async_to_lds_b32)
#define HAVE_ASYNC_LDS 1
#endif
#endif
#ifndef HAVE_ASYNC_LDS
#define HAVE_ASYNC_LDS 0
#endif

#if defined(__has_builtin)
#if __has_builtin(__builtin_amdgcn_s_wait_asynccnt)
#define WAIT_ASYNCCNT(n) __builtin_amdgcn_s_wait_asynccnt(n)
#endif
#endif
#ifndef WAIT_ASYNCCNT
#define WAIT_ASYNCCNT(n) asm volatile("s_wait_asynccnt %0" ::"i"(n) : "memory")
#endif

// wave32 lane broadcast via v_readlane_b32 (uniform result -> SGPR)
__device__ __forceinline__ int   bcast_i(int v, int lane)  {
    return __builtin_amdgcn_readlane(v, lane);
}
__device__ __forceinline__ float bcast_f(float v, int lane) {
    return __int_as_float(__builtin_amdgcn_readlane(__float_as_int(v), lane));
}

// process 32 edges whose metadata sits in (sv, dv, wv) across the wave's lanes
__device__ __forceinline__ void process_batch32(const float* __restrict__ x,
                                                float* __restrict__ y,
                                                int sv, int dv, float wv,
                                                int lane) {
#pragma unroll
    for (int j = 0; j < 32; ++j) {
        const int   s  = bcast_i(sv, j);   // SGPR row index
        const int   d  = bcast_i(dv, j);   // SGPR row index
        const float we = bcast_f(wv, j);   // SGPR weight
        const v4f v =
            reinterpret_cast<const v4f*>(x + (size_t)s * D_FEAT)[lane];
        float* yr = y + (size_t)d * D_FEAT + lane * 4;
        atomicAdd(yr + 0, v.x * we);       // no-return global_atomic_add_f32
        atomicAdd(yr + 1, v.y * we);
        atomicAdd(yr + 2, v.z * we);
        atomicAdd(yr + 3, v.w * we);
    }
}

// ---------------------------------------------------------------------------
// y = x  (vectorized 16B copy; residual-add initialization for a hop)
// ---------------------------------------------------------------------------
__global__ void copy_f4_kernel(const v4f* __restrict__ src,
                               v4f* __restrict__ dst, int n4) {
    int i = blockIdx.x * blockDim.x + threadIdx.x;
    const int stride = gridDim.x * blockDim.x;
    for (; i < n4; i += stride) {
        dst[i] = src[i];
    }
}

// ---------------------------------------------------------------------------
// For each edge e: y[dst[e]] += x[src[e]] * w[e]
//
// Wave32 batched scheme with async double-buffered metadata staging:
//  * each wave owns a private LDS slot (2 x {src[32], dst[32], w[32]})
//  * while processing batch t, GLOBAL_LOAD_ASYNC_TO_LDS_B32 pulls batch t+1
//    (tracked by ASYNCcnt; s_wait_asynccnt before consuming)
//  * per edge: v_readlane broadcast of (s, d, w) -> SGPRs, one coalesced
//    512B row gather (lane L handles floats [4L,4L+4)), 4 no-return
//    global_atomic_add_f32 per lane into the destination row
// All node data is L2-resident (51.2MB << 192MB L2).
// ---------------------------------------------------------------------------
__global__ void __launch_bounds__(256)
edge_scatter_kernel(const float* __restrict__ x,
                    const float* __restrict__ w,
                    const int* __restrict__ src,
                    const int* __restrict__ dst,
                    float* __restrict__ y,
                    int n_edges) {
    const int lane   = threadIdx.x & 31;
    const int wslot  = threadIdx.x >> 5;                       // wave in block
    const int wave   = (blockIdx.x * blockDim.x + threadIdx.x) >> 5;
    const int nwaves = (gridDim.x * blockDim.x) >> 5;
    const int step   = nwaves * 32;

    int base = wave * 32;

#if HAVE_ASYNC_LDS
    // per-wave double buffer: [buf][0..31]=src [32..63]=dst [64..95]=w bits
    __shared__ int smem[8 * 2 * 96];
    int* const slot = &smem[wslot * 192];

    // stage batch 0
    if (base + 32 <= n_edges) {
        const int e = base + lane;
        __builtin_amdgcn_global_load_async_to_lds_b32(
            (int*)(src + e), slot + lane, 0, 0);
        __builtin_amdgcn_global_load_async_to_lds_b32(
            (int*)(dst + e), slot + 32 + lane, 0, 0);
        __builtin_amdgcn_global_load_async_to_lds_b32(
            (int*)((const int*)w + e), slot + 64 + lane, 0, 0);
    }

    int b = 0;
    for (; base + 32 <= n_edges; base += step, b ^= 1) {
        int* const cur = slot + b * 96;
        int* const nxt = slot + (b ^ 1) * 96;

        const int nbase = base + step;
        const bool have_next = (nbase + 32 <= n_edges);
        if (have_next) {
            const int e = nbase + lane;
            __builtin_amdgcn_global_load_async_to_lds_b32(
                (int*)(src + e), nxt + lane, 0, 0);
            __builtin_amdgcn_global_load_async_to_lds_b32(
                (int*)(dst + e), nxt + 32 + lane, 0, 0);
            __builtin_amdgcn_global_load_async_to_lds_b32(
                (int*)((const int*)w + e), nxt + 64 + lane, 0, 0);
            WAIT_ASYNCCNT(3);   // batch t complete (in-order), t+1 in flight
        } else {
            WAIT_ASYNCCNT(0);
        }

        const int   sv = cur[lane];                     // ds_load_b32
        const int   dv = cur[32 + lane];
        const float wv = __int_as_float(cur[64 + lane]);

        // warm this lane's edge's source row (512B) into near caches
        const char* row = (const char*)(x + (size_t)sv * D_FEAT);
        __builtin_prefetch(row, 0, 3);
        __builtin_prefetch(row + 256, 0, 3);

        process_batch32(x, y, sv, dv, wv, lane);
    }
#else
    for (; base + 32 <= n_edges; base += step) {
        const int e = base + lane;
        const int   sv = __builtin_nontemporal_load(&src[e]);
        const int   dv = __builtin_nontemporal_load(&dst[e]);
        const float wv = __builtin_nontemporal_load(&w[e]);
        const char* row = (const char*)(x + (size_t)sv * D_FEAT);
        __builtin_prefetch(row, 0, 3);
        __builtin_prefetch(row + 256, 0, 3);
        process_batch32(x, y, sv, dv, wv, lane);
    }
#endif

    // tail: at most one partial batch for one wave
    if (base < n_edges) {
        const int e = base + lane;
        int   sv = 0, dv = 0;
        float wv = 0.0f;
        if (e < n_edges) {
            sv = src[e];
            dv = dst[e];
            wv = w[e];
        }
        const int cnt = n_edges - base;
        for (int j = 0; j < cnt; ++j) {
            const int   s  = bcast_i(sv, j);
            const int   d  = bcast_i(dv, j);
            const float we = bcast_f(wv, j);
            const v4f v =
                reinterpret_cast<const v4f*>(x + (size_t)s * D_FEAT)[lane];
            float* yr = y + (size_t)d * D_FEAT + lane * 4;
            atomicAdd(yr + 0, v.x * we);
            atomicAdd(yr + 1, v.y * we);
            atomicAdd(yr + 2, v.z * we);
            atomicAdd(yr + 3, v.w * we);
        }
    }
}

// ---------------------------------------------------------------------------
// out[r] = y[sel[r]]  — one wave per selected row; NT stores (never re-read)
// ---------------------------------------------------------------------------
__global__ void gather_rows_kernel(const float* __restrict__ y,
                                   const int* __restrict__ sel,
                                   float* __restrict__ out, int n_pool) {
    const int lane   = threadIdx.x & 31;
    int wave         = (blockIdx.x * blockDim.x + threadIdx.x) >> 5;
    const int nwaves = (gridDim.x * blockDim.x) >> 5;

    for (int r = wave; r < n_pool; r += nwaves) {
        const int s = sel[r];
        const v4f v =
            reinterpret_cast<const v4f*>(y + (size_t)s * D_FEAT)[lane];
        __builtin_nontemporal_store(
            v, &reinterpret_cast<v4f*>(out + (size_t)r * D_FEAT)[lane]);
    }
}

// ---------------------------------------------------------------------------
// Launch: ping-pong two N*128 float buffers in d_ws.
//   b0 = x; scatter(x -> b0); b1 = b0; scatter(b0 -> b1); out = b1[sel]
// ---------------------------------------------------------------------------
extern "C" void kernel_launch(void* const* d_in, const int* in_sizes, int n_in,
                              void* d_out, int out_size, void* d_ws, size_t ws_size,
                              hipStream_t stream) {
    const float* x   = (const float*)d_in[0];   // [N, 128] f32
    const float* w   = (const float*)d_in[1];   // [E]      f32
    const int*   ei  = (const int*)  d_in[2];   // [2, E]   int
    const int*   sel = (const int*)  d_in[3];   // [P]      int

    const int n_nodes = in_sizes[0] / D_FEAT;
    const int n_edges = in_sizes[2] / 2;
    const int n_pool  = in_sizes[3];

    const int* src = ei;             // edge_index[0]
    const int* dst = ei + n_edges;   // edge_index[1]

    float* b0 = (float*)d_ws;
    float* b1 = b0 + (size_t)n_nodes * D_FEAT;

    const int n4 = (n_nodes * D_FEAT) / 4;
    int copy_blocks = (n4 + 255) / 256;
    if (copy_blocks > 16384) copy_blocks = 16384;

    // 2048 blocks * 8 waves = 16384 waves; ~3 batches of 32 edges per wave
    const int edge_blocks   = 2048;
    const int gather_blocks = (n_pool * 32 + 255) / 256;

    // hop 1: b0 = x + scatter(x)
    copy_f4_kernel<<<copy_blocks, 256, 0, stream>>>(
        (const v4f*)x, (v4f*)b0, n4);
    edge_scatter_kernel<<<edge_blocks, 256, 0, stream>>>(
        x, w, src, dst, b0, n_edges);

    // hop 2: b1 = b0 + scatter(b0)
    copy_f4_kernel<<<copy_blocks, 256, 0, stream>>>(
        (const v4f*)b0, (v4f*)b1, n4);
    edge_scatter_kernel<<<edge_blocks, 256, 0, stream>>>(
        b0, w, src, dst, b1, n_edges);

    // final k-MIS gather
    gather_rows_kernel<<<gather_blocks, 256, 0, stream>>>(
        b1, sel, (float*)d_out, n_pool);
}